// SubwordStackLSTMCell_57930518888543
// MI455X (gfx1250) — compile-verified
//
#include <hip/hip_runtime.h>
#include <hip/hip_bf16.h>
#include <cstdint>

#define Hh 256
#define Ii 256
#define Bb 64
#define Ww 64
#define Ss 65
#define STACK_ELEMS ((size_t)Bb * Ww * Ss * Hh)   // 68,157,440 floats per stack tensor

typedef __attribute__((ext_vector_type(2))) float v2f;
typedef __attribute__((ext_vector_type(4))) float v4f;
typedef __attribute__((ext_vector_type(8))) float v8f;

__device__ __forceinline__ v8f wmma4(v2f a, v2f b, v8f c) {
    // V_WMMA_F32_16X16X4_F32 : exact fp32 MMA, D = A(16x4) x B(4x16) + C(16x16)
    return __builtin_amdgcn_wmma_f32_16x16x4_f32(false, a, false, b, (short)0, c, false, false);
}

__device__ __forceinline__ float sigm(float x) { return 1.0f / (1.0f + __expf(-x)); }

// ---------------- Kernel 1: HBM-saturating non-temporal float4 copy ----------------
__global__ void nt_copy4(const v4f* __restrict__ src, v4f* __restrict__ dst, long n4) {
    long i = (long)blockIdx.x * blockDim.x + threadIdx.x;
    long stride = (long)gridDim.x * blockDim.x;
    for (; i < n4; i += stride) {
        v4f v = __builtin_nontemporal_load(src + i);
        __builtin_nontemporal_store(v, dst + i);
    }
}

// ---------------- Kernel 2: gather h,c at [b, pos_word, pos_subword] ----------------
__global__ void gather_hc(const float* __restrict__ sh, const float* __restrict__ sc,
                          const int* __restrict__ pw, const int* __restrict__ ps,
                          float* __restrict__ hbuf, float* __restrict__ cbuf) {
    int b = blockIdx.x, j = threadIdx.x;
    size_t off = (((size_t)b * Ww + pw[b]) * Ss + ps[b]) * Hh + j;
    hbuf[b * Hh + j] = sh[off];
    cbuf[b * Hh + j] = sc[off];
}

// ---------------- Kernel 3: fused gates GEMM via V_WMMA_F32_16X16X4_F32 ----------------
// gates (64 x 2048):
//   cols [0,1024)    : char @ W_ih_r^T + h @ W_hh_r^T + b_ih_r + b_hh_r
//   cols [1024,2048) : char @ W_ih_l^T + b_ih_l + b_hh_l   (zero initial state)
// One wave per 16x16 tile; 512 tiles total.
__global__ void gates_wmma(const float* __restrict__ x,     // char 64x256
                           const float* __restrict__ hbuf,  // gathered h 64x256
                           const float* __restrict__ Wih_r, const float* __restrict__ Whh_r,
                           const float* __restrict__ bih_r, const float* __restrict__ bhh_r,
                           const float* __restrict__ Wih_l,
                           const float* __restrict__ bih_l, const float* __restrict__ bhh_l,
                           float* __restrict__ gates) {
    int tile = blockIdx.x;        // 0..511
    int mb = (tile & 3) * 16;     // batch-row base
    int cb = (tile >> 2) * 16;    // combined gate-column base (0..2047)
    int lane = threadIdx.x;       // 0..31, EXEC all ones
    int ln = lane & 15;           // A-row / B-col within tile
    int kb = (lane >> 4) << 1;    // K sub-offset {0,2}
    int hi = (lane >> 4) * 8;     // C/D row split

    const float* xrow = x + (size_t)(mb + ln) * Ii;
    v8f acc = {};
    float bias;
    if (cb < 1024) {
        const float* hrow = hbuf + (size_t)(mb + ln) * Hh;
        const float* wi = Wih_r + (size_t)(cb + ln) * Ii;   // row-major (N,K): contiguous in K
        const float* wh = Whh_r + (size_t)(cb + ln) * Hh;
        for (int k = 0; k < Ii; k += 4) {
            v2f a0 = *(const v2f*)(xrow + k + kb);
            v2f w0 = *(const v2f*)(wi + k + kb);
            acc = wmma4(a0, w0, acc);
            v2f a1 = *(const v2f*)(hrow + k + kb);
            v2f w1 = *(const v2f*)(wh + k + kb);
            acc = wmma4(a1, w1, acc);
        }
        bias = bih_r[cb + ln] + bhh_r[cb + ln];
    } else {
        int c2 = cb - 1024;
        const float* wi = Wih_l + (size_t)(c2 + ln) * Ii;
        for (int k = 0; k < Ii; k += 4) {
            v2f a0 = *(const v2f*)(xrow + k + kb);
            v2f w0 = *(const v2f*)(wi + k + kb);
            acc = wmma4(a0, w0, acc);
        }
        bias = bih_l[c2 + ln] + bhh_l[c2 + ln];
    }
    // C/D layout: vgpr v -> row mb + hi + v, col cb + ln
    for (int v = 0; v < 8; ++v)
        gates[(size_t)(mb + hi + v) * 2048 + cb + ln] = acc[v] + bias;
}

// ---------------- Kernel 4: LSTM nonlinearity + scatter + concat ----------------
__global__ void lstm_pointwise(const float* __restrict__ gates, const float* __restrict__ cbuf,
                               const int* __restrict__ pw, const int* __restrict__ ps,
                               float* __restrict__ out_sh, float* __restrict__ out_sc,
                               float* __restrict__ hcat) {
    int b = blockIdx.x, j = threadIdx.x;
    const float* g = gates + (size_t)b * 2048;
    // right cell (gate order i,f,g,o)
    float ig = sigm(g[j]);
    float fg = sigm(g[256 + j]);
    float gg = tanhf(g[512 + j]);
    float og = sigm(g[768 + j]);
    float cn = fg * cbuf[b * Hh + j] + ig * gg;
    float hr = og * tanhf(cn);
    size_t off = (((size_t)b * Ww + pw[b]) * Ss + (ps[b] + 1)) * Hh + j;
    out_sh[off] = hr;
    out_sc[off] = cn;
    // left cell with zero state: c = sigmoid(i)*tanh(g)
    float il = sigm(g[1024 + j]);
    float gl = tanhf(g[1536 + j]);
    float ol = sigm(g[1792 + j]);
    float hl = ol * tanhf(il * gl);
    hcat[(size_t)b * 512 + j] = hr;
    hcat[(size_t)b * 512 + 256 + j] = hl;
}

// ---------------- Kernel 5: compose GEMM (64x512 @ 512x256) + tanh ----------------
__global__ void compose_wmma(const float* __restrict__ hcat, const float* __restrict__ Wc,
                             const float* __restrict__ bc, float* __restrict__ out) {
    int tile = blockIdx.x;        // 0..63
    int mb = (tile & 3) * 16;
    int cb = (tile >> 2) * 16;    // 0..255
    int lane = threadIdx.x;
    int ln = lane & 15;
    int kb = (lane >> 4) << 1;
    int hi = (lane >> 4) * 8;

    const float* arow = hcat + (size_t)(mb + ln) * 512;
    const float* wrow = Wc + (size_t)(cb + ln) * 512;
    v8f acc = {};
    for (int k = 0; k < 512; k += 4) {
        v2f a = *(const v2f*)(arow + k + kb);
        v2f w = *(const v2f*)(wrow + k + kb);
        acc = wmma4(a, w, acc);
    }
    float bias = bc[cb + ln];
    for (int v = 0; v < 8; ++v)
        out[(size_t)(mb + hi + v) * Hh + cb + ln] = tanhf(acc[v] + bias);
}

extern "C" void kernel_launch(void* const* d_in, const int* in_sizes, int n_in,
                              void* d_out, int out_size, void* d_ws, size_t ws_size,
                              hipStream_t stream) {
    const float* xc    = (const float*)d_in[0];   // char (64,256)
    const float* sh    = (const float*)d_in[1];   // stack_hidden
    const float* sc    = (const float*)d_in[2];   // stack_cell
    const int*   pw    = (const int*)d_in[3];     // pos_word
    const int*   ps    = (const int*)d_in[4];     // pos_subword
    const float* Wih_r = (const float*)d_in[5];
    const float* Whh_r = (const float*)d_in[6];
    const float* bih_r = (const float*)d_in[7];
    const float* bhh_r = (const float*)d_in[8];
    const float* Wih_l = (const float*)d_in[9];
    const float* bih_l = (const float*)d_in[11];
    const float* bhh_l = (const float*)d_in[12];
    const float* Wc    = (const float*)d_in[13];
    const float* bc    = (const float*)d_in[14];

    float* out     = (float*)d_out;
    float* out_sub = out;                         // (64,256)
    float* out_sh  = out + (size_t)Bb * Hh;       // (64,64,65,256)
    float* out_sc  = out_sh + STACK_ELEMS;        // (64,64,65,256)

    float* ws    = (float*)d_ws;
    float* hbuf  = ws;                 // 16384 floats
    float* cbuf  = ws + 16384;         // 16384 floats
    float* gates = ws + 32768;         // 64*2048 = 131072 floats
    float* hcat  = ws + 163840;        // 64*512  = 32768 floats

    long n4 = (long)(STACK_ELEMS / 4);
    nt_copy4<<<4096, 256, 0, stream>>>((const v4f*)sh, (v4f*)out_sh, n4);
    nt_copy4<<<4096, 256, 0, stream>>>((const v4f*)sc, (v4f*)out_sc, n4);
    gather_hc<<<Bb, Hh, 0, stream>>>(sh, sc, pw, ps, hbuf, cbuf);
    gates_wmma<<<512, 32, 0, stream>>>(xc, hbuf, Wih_r, Whh_r, bih_r, bhh_r,
                                       Wih_l, bih_l, bhh_l, gates);
    lstm_pointwise<<<Bb, Hh, 0, stream>>>(gates, cbuf, pw, ps, out_sh, out_sc, hcat);
    compose_wmma<<<64, 32, 0, stream>>>(hcat, Wc, bc, out_sub);
}